// RGate_72773925863426
// MI455X (gfx1250) — compile-verified
//
#include <hip/hip_runtime.h>
#include <hip/hip_bf16.h>

// CDNA5 / gfx1250: wave32, WMMA f32 16x16x4.
typedef __attribute__((ext_vector_type(2))) float v2f;
typedef __attribute__((ext_vector_type(8))) float v8f;

#define CB_PER_WAVE 4                  // 16-fiber column blocks per wave
#define TOTAL_CB4   (1 << 14)          // 2^18 fibers / 16 per column block
#define WAVES4      (TOTAL_CB4 / CB_PER_WAVE)   // 4096
#define BLOCKS4     (WAVES4 / 8)                // 512 blocks of 8 waves

static __device__ __forceinline__ v8f wmma4(v2f a, v2f b, v8f c) {
  // D(16x16,f32) += A(16x4,f32) * B(4x16,f32) + C
  return __builtin_amdgcn_wmma_f32_16x16x4_f32(
      /*neg_a=*/false, a, /*neg_b=*/false, b,
      /*c_mod=*/(short)0, c, /*reuse_a=*/false, /*reuse_b=*/false);
}

// 32x32 real representation of the 4-qubit gate G = kron of 4 single-qubit
// rotations on global bits qbase..qbase+3. Axis i of the reference reshape is
// bit (21-i), so bit (qbase+j) uses angle[21-(qbase+j)].
// W = [[Gr, -Gi],[Gi, Gr]], G[t,t'] = (-i)^h * prod_j (c_j or s_j),
// h = popcount(t ^ t').
static __device__ __forceinline__ void fill_W32(float* Wlds,
                                                const float* __restrict__ angle,
                                                int qbase) {
  float c[4], s[4];
#pragma unroll
  for (int j = 0; j < 4; ++j) {
    float a = 0.5f * angle[21 - (qbase + j)];
    c[j] = cosf(a);
    s[j] = sinf(a);
  }
#pragma unroll
  for (int e = threadIdx.x; e < 1024; e += 256) {
    int r = e >> 5, k = e & 31;
    int d = (r & 15) ^ (k & 15);
    int h = __popc(d);
    float P = 1.0f;
#pragma unroll
    for (int j = 0; j < 4; ++j) P *= ((d >> j) & 1) ? s[j] : c[j];
    // (-i)^h: 0->+1, 1->-i, 2->-1, 3->+i, 4->+1
    float Gr = ((h & 1) == 0) ? (((h & 3) == 0) ? P : -P) : 0.0f;
    float Gi = ((h & 1) == 1) ? (((h & 3) == 1) ? -P : P) : 0.0f;
    float w;
    if (r < 16) w = (k < 16) ? Gr : -Gi;
    else        w = (k < 16) ? Gi :  Gr;
    Wlds[r * 32 + k] = w;
  }
  __syncthreads();
}

// Pass 0: bits 0..3, real input x -> complex psi. Imag(input)=0, so only the
// Re K-chunks (k=0..15, s=0..3) contribute: 8 WMMAs per column block.
__global__ void RGate_pass4_0(const float* __restrict__ x,
                              float2* __restrict__ psi,
                              const float* __restrict__ angle) {
  __shared__ float Wlds[1024];
  fill_W32(Wlds, angle, 0);

  const int lane = threadIdx.x & 31;
  const int m    = lane & 15;
  const int hi   = lane >> 4;          // 0 for lanes 0-15, 1 for lanes 16-31
  const int koff = hi << 1;

  v2f A0[4], A1[4];                    // row tiles 0..15 / 16..31, Re chunks
#pragma unroll
  for (int s = 0; s < 4; ++s) {
    A0[s].x = Wlds[m * 32 + 4 * s + koff];
    A0[s].y = Wlds[m * 32 + 4 * s + koff + 1];
    A1[s].x = Wlds[(m + 16) * 32 + 4 * s + koff];
    A1[s].y = Wlds[(m + 16) * 32 + 4 * s + koff + 1];
  }

  const int wave = blockIdx.x * (blockDim.x >> 5) + (threadIdx.x >> 5);
#pragma unroll 1
  for (int i = 0; i < CB_PER_WAVE; ++i) {
    const int cb   = wave * CB_PER_WAVE + i;
    const int n    = cb * 16 + m;      // fiber id; q=0 -> element = 16n + t
    const int base = n << 4;

    float xv[8];
#pragma unroll
    for (int p = 0; p < 8; ++p) {
      int t = (p >> 1) * 4 + koff + (p & 1);  // lanes<16: {0,1,4,5,8,9,12,13}
      xv[p] = x[base + t];
    }

    v8f d0 = {}, d1 = {};
#pragma unroll
    for (int u = 0; u < 4; ++u) {
      v2f Br; Br.x = xv[2 * u]; Br.y = xv[2 * u + 1];
      d0 = wmma4(A0[u], Br, d0);
      d1 = wmma4(A1[u], Br, d1);
    }

    const int tbase = hi << 3;         // lanes<16 own t=0..7, lanes>=16 t=8..15
#pragma unroll
    for (int j = 0; j < 8; ++j)
      psi[base + tbase + j] = make_float2(d0[j], d1[j]);   // (Re, Im), no shfl
  }
}

// Bits [qbase, qbase+3], qbase in {4,8,12,16}; in-place complex update.
__global__ void RGate_pass4(float2* __restrict__ psi,
                            const float* __restrict__ angle,
                            int qbase) {
  __shared__ float Wlds[1024];
  fill_W32(Wlds, angle, qbase);

  const int lane = threadIdx.x & 31;
  const int m    = lane & 15;
  const int hi   = lane >> 4;
  const int koff = hi << 1;

  v2f A0[8], A1[8];                    // 8 K-chunks x 2 row tiles
#pragma unroll
  for (int s = 0; s < 8; ++s) {
    A0[s].x = Wlds[m * 32 + 4 * s + koff];
    A0[s].y = Wlds[m * 32 + 4 * s + koff + 1];
    A1[s].x = Wlds[(m + 16) * 32 + 4 * s + koff];
    A1[s].y = Wlds[(m + 16) * 32 + 4 * s + koff + 1];
  }

  const int st   = 1 << qbase;
  const int mask = st - 1;
  const int wave = blockIdx.x * (blockDim.x >> 5) + (threadIdx.x >> 5);
#pragma unroll 1
  for (int i = 0; i < CB_PER_WAVE; ++i) {
    const int cb   = wave * CB_PER_WAVE + i;
    const int n    = cb * 16 + m;
    const int base = ((n >> qbase) << (qbase + 4)) | (n & mask);  // t=0 element

    float2 z[8];
#pragma unroll
    for (int p = 0; p < 8; ++p) {
      int t = (p >> 1) * 4 + koff + (p & 1);
      z[p] = psi[base + t * st];
    }

    v8f d0 = {}, d1 = {};
#pragma unroll
    for (int u = 0; u < 4; ++u) {       // Re rows of V (k = 0..15)
      v2f Br; Br.x = z[2 * u].x; Br.y = z[2 * u + 1].x;
      d0 = wmma4(A0[u], Br, d0);
      d1 = wmma4(A1[u], Br, d1);
    }
#pragma unroll
    for (int u = 0; u < 4; ++u) {       // Im rows of V (k = 16..31)
      v2f Bi; Bi.x = z[2 * u].y; Bi.y = z[2 * u + 1].y;
      d0 = wmma4(A0[4 + u], Bi, d0);
      d1 = wmma4(A1[4 + u], Bi, d1);
    }

    const int tbase = hi << 3;
#pragma unroll
    for (int j = 0; j < 8; ++j)
      psi[base + (tbase + j) * st] = make_float2(d0[j], d1[j]);
  }
}

// Top two bits: bit 20 (angle[1]) and bit 21 (angle[0]); VALU butterflies.
__global__ void RGate_top2(float2* __restrict__ psi,
                           const float* __restrict__ angle) {
  const int r  = blockIdx.x * blockDim.x + threadIdx.x;   // 0 .. 2^20-1
  const int st = 1 << 20;
  const float a1 = 0.5f * angle[1], a0 = 0.5f * angle[0];
  const float c1 = cosf(a1), s1 = sinf(a1);
  const float c0 = cosf(a0), s0 = sinf(a0);

  float2 z0 = psi[r];
  float2 z1 = psi[r + st];
  float2 z2 = psi[r + 2 * st];
  float2 z3 = psi[r + 3 * st];

  // bit 20: pairs (0,1) and (2,3):  w = c*z - i*s*partner
  float2 y0 = make_float2(c1 * z0.x + s1 * z1.y, c1 * z0.y - s1 * z1.x);
  float2 y1 = make_float2(c1 * z1.x + s1 * z0.y, c1 * z1.y - s1 * z0.x);
  float2 y2 = make_float2(c1 * z2.x + s1 * z3.y, c1 * z2.y - s1 * z3.x);
  float2 y3 = make_float2(c1 * z3.x + s1 * z2.y, c1 * z3.y - s1 * z2.x);
  // bit 21: pairs (0,2) and (1,3)
  float2 o0 = make_float2(c0 * y0.x + s0 * y2.y, c0 * y0.y - s0 * y2.x);
  float2 o2 = make_float2(c0 * y2.x + s0 * y0.y, c0 * y2.y - s0 * y0.x);
  float2 o1 = make_float2(c0 * y1.x + s0 * y3.y, c0 * y1.y - s0 * y3.x);
  float2 o3 = make_float2(c0 * y3.x + s0 * y1.y, c0 * y3.y - s0 * y1.x);

  psi[r]          = o0;
  psi[r + st]     = o1;
  psi[r + 2 * st] = o2;
  psi[r + 3 * st] = o3;
}

extern "C" void kernel_launch(void* const* d_in, const int* in_sizes, int n_in,
                              void* d_out, int out_size, void* d_ws, size_t ws_size,
                              hipStream_t stream) {
  (void)in_sizes; (void)n_in; (void)out_size; (void)d_ws; (void)ws_size;
  const float* x     = (const float*)d_in[0];   // [2^22] float32
  const float* angle = (const float*)d_in[1];   // [22] float32
  float2* psi        = (float2*)d_out;          // [2^22] complex64 as float2

  RGate_pass4_0<<<BLOCKS4, 256, 0, stream>>>(x, psi, angle);
  for (int q = 4; q <= 16; q += 4)
    RGate_pass4<<<BLOCKS4, 256, 0, stream>>>(psi, angle, q);
  RGate_top2<<<(1 << 20) / 256, 256, 0, stream>>>(psi, angle);
}